// VoxelPooling_24910810317274
// MI455X (gfx1250) — compile-verified
//
#include <hip/hip_runtime.h>
#include <stdint.h>

#define VOX_C 32
#define ROW_PAD 33                 // 32 + 1 pad DWORD inserted by TDM -> conflict-free LDS
#define GRID_G (1u << 21)          // 128*128*128 linear voxel ids (covers STRIDE=128 layout)
#define SCAN_BLK 256
#define SCAN_ELEMS 1024            // 4 per thread

// ---- monotonic uint encoding of float (order-preserving; 0u is identity for max) ----
__device__ __forceinline__ unsigned enc_f32(float f) {
  unsigned u = __float_as_uint(f);
  return (u & 0x80000000u) ? ~u : (u | 0x80000000u);
}
__device__ __forceinline__ float dec_f32(unsigned e) {
  unsigned u = (e & 0x80000000u) ? (e & 0x7FFFFFFFu) : ~e;
  return __uint_as_float(u);
}

// ---- K1: zero encoded feature region, flags, uniq ----
__global__ void k_init(unsigned* __restrict__ featU, unsigned* __restrict__ flags,
                       int* __restrict__ uniq, int nFeat, int nPts) {
  int i = blockIdx.x * blockDim.x + threadIdx.x;
  if (i < nFeat) featU[i] = 0u;
  if (i < (int)GRID_G) flags[i] = 0u;
  if (i < nPts) uniq[i] = -1;
}

// ---- K2: occupancy flags ----
__global__ void k_mark(const float* __restrict__ pos, const float* __restrict__ vs,
                       unsigned* __restrict__ flags, int n) {
  int i = blockIdx.x * blockDim.x + threadIdx.x;
  if (i >= n) return;
  float v = vs[0];
  int cx = (int)floorf(pos[3 * i + 0] / v);
  int cy = (int)floorf(pos[3 * i + 1] / v);
  int cz = (int)floorf(pos[3 * i + 2] / v);
  unsigned lin = ((unsigned)cx + 128u * (unsigned)cy + 16384u * (unsigned)cz) & (GRID_G - 1u);
  flags[lin] = 1u;
}

// ---- K3a: per-block reduction of flags ----
__global__ __launch_bounds__(SCAN_BLK) void k_scan_reduce(const unsigned* __restrict__ flags,
                                                          unsigned* __restrict__ bsum) {
  __shared__ unsigned s[SCAN_BLK];
  int b = blockIdx.x, t = threadIdx.x;
  const uint4* p = (const uint4*)(flags + (size_t)b * SCAN_ELEMS);
  uint4 v = p[t];
  s[t] = v.x + v.y + v.z + v.w;
  __syncthreads();
  for (int off = SCAN_BLK / 2; off > 0; off >>= 1) {
    if (t < off) s[t] += s[t + off];
    __syncthreads();
  }
  if (t == 0) bsum[b] = s[0];
}

// ---- K3b: exclusive scan of block sums (tiny, single thread is fine) ----
__global__ void k_scan_bsum(unsigned* __restrict__ bsum, int nb) {
  if (blockIdx.x == 0 && threadIdx.x == 0) {
    unsigned acc = 0;
    for (int i = 0; i < nb; ++i) { unsigned v = bsum[i]; bsum[i] = acc; acc += v; }
  }
}

// ---- K3c: per-block exclusive scan -> ranks ----
__global__ __launch_bounds__(SCAN_BLK) void k_scan_write(const unsigned* __restrict__ flags,
                                                         const unsigned* __restrict__ bsum,
                                                         unsigned* __restrict__ ranks) {
  __shared__ unsigned s[SCAN_BLK];
  int b = blockIdx.x, t = threadIdx.x;
  const uint4* p = (const uint4*)(flags + (size_t)b * SCAN_ELEMS);
  uint4 v = p[t];
  unsigned mySum = v.x + v.y + v.z + v.w;
  s[t] = mySum;
  __syncthreads();
  for (int off = 1; off < SCAN_BLK; off <<= 1) {   // inclusive Hillis-Steele
    unsigned add = (t >= off) ? s[t - off] : 0u;
    __syncthreads();
    s[t] += add;
    __syncthreads();
  }
  unsigned base = bsum[b] + (t ? s[t - 1] : 0u);
  unsigned idx0 = (unsigned)b * SCAN_ELEMS + (unsigned)t * 4u;
  unsigned r = base;
  ranks[idx0 + 0] = r; r += v.x;
  ranks[idx0 + 1] = r; r += v.y;
  ranks[idx0 + 2] = r; r += v.z;
  ranks[idx0 + 3] = r;
}

// ---- K4: compact sorted-unique voxel ids ----
__global__ void k_compact(const unsigned* __restrict__ flags, const unsigned* __restrict__ ranks,
                          int* __restrict__ uniq, int nPts) {
  int g = blockIdx.x * blockDim.x + threadIdx.x;
  if (g >= (int)GRID_G) return;
  if (flags[g]) {
    unsigned r = ranks[g];
    if (r < (unsigned)nPts) uniq[r] = g;
  }
}

// ---- K5: TDM-staged scatter: features+positions -> LDS via tensor DMA, atomic-max scatter ----
typedef __attribute__((ext_vector_type(4))) unsigned int tdm_v4u;
typedef __attribute__((ext_vector_type(8))) int tdm_v8i;
typedef __attribute__((ext_vector_type(4))) int tdm_v4i;

#define LDS_FEAT_WORDS (256 * ROW_PAD)   // 33792 B: 256 rows, stride 33 (TDM pad)
#define LDS_POS_WORDS (256 * 3)          // 3072 B

__global__ __launch_bounds__(256) void k_scatter(const float* __restrict__ pos,
                                                 const float* __restrict__ feat,
                                                 const float* __restrict__ vs,
                                                 const unsigned* __restrict__ ranks,
                                                 unsigned* __restrict__ outFeatU, int n) {
  __shared__ float smem[LDS_FEAT_WORDS + LDS_POS_WORDS];
  float* ldsFeat = smem;                               // LDS byte offset 0
  float* ldsPos = smem + LDS_FEAT_WORDS;               // LDS byte offset 33792
  const int t = threadIdx.x;
  const int blockBase = blockIdx.x * 256;
  const int waveId = t >> 5;
  const int waveBase = blockBase + (waveId << 5);
  const int remW = n - waveBase;                       // wave-uniform

#if defined(__has_builtin) && __has_builtin(__builtin_amdgcn_tensor_load_to_lds) && __has_builtin(__builtin_amdgcn_s_wait_tensorcnt)
  if (remW > 0) {
    unsigned rows = (unsigned)(remW < 32 ? remW : 32); // tensor_dim1 < tile_dim1 -> OOB rows read 0
    tdm_v4i gz = {0, 0, 0, 0};
    tdm_v8i g5 = {0, 0, 0, 0, 0, 0, 0, 0};

    // --- descriptor A: 32x32 f32 feature tile, TDM-inserted 1-DWORD row pad (stride 33) ---
    {
      unsigned long long ga = (unsigned long long)(const void*)(feat + (size_t)waveBase * VOX_C);
      tdm_v4u g0;
      g0[0] = 1u;                                      // count=1 valid descriptor
      g0[1] = (unsigned)(waveId * (32 * ROW_PAD * 4)); // lds_addr: per-wave 4224B sub-tile
      g0[2] = (unsigned)(ga & 0xFFFFFFFFull);
      g0[3] = (unsigned)((ga >> 32) & 0x1FFFFFFull) | (2u << 30);  // addr[56:32] | type=2
      tdm_v8i g1;
      g1[0] = (int)((2u << 16) | (1u << 20) | (4u << 22)); // data_size=4B, pad_en, interval=32DW, amount=1DW
      g1[1] = (int)((unsigned)VOX_C << 16);            // tensor_dim0 = 32
      g1[2] = (int)(rows << 16);                       // tensor_dim1 = rows
      g1[3] = (int)(32u << 16);                        // tile_dim0 = 32
      g1[4] = (int)32u;                                // tile_dim1 = 32, tile_dim2 = 0
      g1[5] = (int)VOX_C;                              // tensor_dim0_stride = 32 elements
      g1[6] = 0;
      g1[7] = 0;
      __builtin_amdgcn_tensor_load_to_lds(g0, g1, gz, gz, g5, 0);
    }
    // --- descriptor B: 32x3 f32 position tile (odd row length -> conflict-free) ---
    {
      unsigned long long ga = (unsigned long long)(const void*)(pos + (size_t)waveBase * 3);
      tdm_v4u g0;
      g0[0] = 1u;
      g0[1] = (unsigned)(LDS_FEAT_WORDS * 4 + waveId * (32 * 3 * 4));
      g0[2] = (unsigned)(ga & 0xFFFFFFFFull);
      g0[3] = (unsigned)((ga >> 32) & 0x1FFFFFFull) | (2u << 30);
      tdm_v8i g1;
      g1[0] = (int)(2u << 16);                         // data_size=4B, no pad
      g1[1] = (int)(3u << 16);                         // tensor_dim0 = 3
      g1[2] = (int)(rows << 16);                       // tensor_dim1 = rows
      g1[3] = (int)(3u << 16);                         // tile_dim0 = 3
      g1[4] = (int)32u;                                // tile_dim1 = 32
      g1[5] = 3;                                       // tensor_dim0_stride = 3 elements
      g1[6] = 0;
      g1[7] = 0;
      __builtin_amdgcn_tensor_load_to_lds(g0, g1, gz, gz, g5, 0);
    }
    __builtin_amdgcn_s_wait_tensorcnt(0);
    asm volatile("" ::: "memory");                     // compiler fence: LDS written by TDM
  }
#else
  // Fallback: stage this thread's own row with plain loads (same padded layout)
  if (blockBase + t < n) {
    const float* src = feat + (size_t)(blockBase + t) * VOX_C;
#pragma unroll
    for (int c = 0; c < VOX_C; ++c) ldsFeat[t * ROW_PAD + c] = src[c];
    const float* sp = pos + (size_t)(blockBase + t) * 3;
#pragma unroll
    for (int k = 0; k < 3; ++k) ldsPos[t * 3 + k] = sp[k];
  }
#endif

  const int p = blockBase + t;
  if (p >= n) return;
  float v = vs[0];
  float px = ldsPos[t * 3 + 0];
  float py = ldsPos[t * 3 + 1];
  float pz = ldsPos[t * 3 + 2];
  int cx = (int)floorf(px / v);
  int cy = (int)floorf(py / v);
  int cz = (int)floorf(pz / v);
  unsigned lin = ((unsigned)cx + 128u * (unsigned)cy + 16384u * (unsigned)cz) & (GRID_G - 1u);
  unsigned row = ranks[lin];
  unsigned* dst = outFeatU + (size_t)row * VOX_C;
  const float* src = &ldsFeat[t * ROW_PAD];            // bank = (t+c)%32 -> conflict-free
#pragma unroll
  for (int c = 0; c < VOX_C; ++c) {
    atomicMax(&dst[c], enc_f32(src[c]));               // order-invariant => deterministic
  }
}

// ---- K6: decode features, write centers + valid, zero padding rows ----
__global__ void k_finalize(const int* __restrict__ uniq, const float* __restrict__ vs,
                           float* __restrict__ outPos, float* __restrict__ outFeat,
                           float* __restrict__ outValid, int n) {
  int r = blockIdx.x * blockDim.x + threadIdx.x;
  if (r >= n) return;
  int g = uniq[r];
  unsigned* fU = (unsigned*)(outFeat + (size_t)r * VOX_C);
  float* fF = outFeat + (size_t)r * VOX_C;
  if (g >= 0) {
    float v = vs[0];
    int cz = g >> 14;
    int rem = g & 16383;
    int cy = rem >> 7;
    int cx = rem & 127;
    outPos[3 * r + 0] = ((float)cx + 0.5f) * v;
    outPos[3 * r + 1] = ((float)cy + 0.5f) * v;
    outPos[3 * r + 2] = ((float)cz + 0.5f) * v;
#pragma unroll
    for (int c = 0; c < VOX_C; ++c) fF[c] = dec_f32(fU[c]);
    outValid[r] = 1.0f;
  } else {
    outPos[3 * r + 0] = 0.0f;
    outPos[3 * r + 1] = 0.0f;
    outPos[3 * r + 2] = 0.0f;
#pragma unroll
    for (int c = 0; c < VOX_C; ++c) fF[c] = 0.0f;
    outValid[r] = 0.0f;
  }
}

extern "C" void kernel_launch(void* const* d_in, const int* in_sizes, int n_in,
                              void* d_out, int out_size, void* d_ws, size_t ws_size,
                              hipStream_t stream) {
  const float* pos = (const float*)d_in[0];
  const float* feat = (const float*)d_in[1];
  const float* vs = (const float*)d_in[2];
  const int n = in_sizes[0] / 3;          // N points
  const int nFeat = n * VOX_C;

  float* outPos = (float*)d_out;
  float* outFeat = outPos + (size_t)3 * n;
  float* outValid = outFeat + (size_t)VOX_C * n;

  unsigned char* ws = (unsigned char*)d_ws;
  unsigned* flags = (unsigned*)ws;                                    // GRID_G u32 (8 MB)
  unsigned* ranks = (unsigned*)(ws + (size_t)GRID_G * 4);             // GRID_G u32 (8 MB)
  unsigned* bsum = (unsigned*)(ws + (size_t)GRID_G * 8);              // GRID_G/1024 u32
  int* uniq = (int*)(ws + (size_t)GRID_G * 8 + (size_t)(GRID_G / SCAN_ELEMS) * 4 + 1024);

  const int nbScan = GRID_G / SCAN_ELEMS;  // 2048
  int initCount = nFeat > (int)GRID_G ? nFeat : (int)GRID_G;
  if (initCount < n) initCount = n;

  k_init<<<(initCount + 255) / 256, 256, 0, stream>>>((unsigned*)outFeat, flags, uniq, nFeat, n);
  k_mark<<<(n + 255) / 256, 256, 0, stream>>>(pos, vs, flags, n);
  k_scan_reduce<<<nbScan, SCAN_BLK, 0, stream>>>(flags, bsum);
  k_scan_bsum<<<1, 1, 0, stream>>>(bsum, nbScan);
  k_scan_write<<<nbScan, SCAN_BLK, 0, stream>>>(flags, bsum, ranks);
  k_compact<<<((int)GRID_G + 255) / 256, 256, 0, stream>>>(flags, ranks, uniq, n);
  k_scatter<<<(n + 255) / 256, 256, 0, stream>>>(pos, feat, vs, ranks, (unsigned*)outFeat, n);
  k_finalize<<<(n + 255) / 256, 256, 0, stream>>>(uniq, vs, outPos, outFeat, outValid, n);
}